// MoLExLinear_18966575579733
// MI455X (gfx1250) — compile-verified
//
#include <hip/hip_runtime.h>
#include <hip/hip_bf16.h>

typedef __attribute__((ext_vector_type(2))) float v2f;
typedef __attribute__((ext_vector_type(8))) float v8f;

#define N_TOK 16384
#define DIM   1024
#define ODIM  1024
#define NEXP  8
#define RANK  8
#define KEXT  64            // E*R extension columns
#define T1W   80            // padded width of t1 = [64 xA | 8 logits | 8 pad]
#define SCALING 2.0f        // alpha/r = 16/8
#define LDSP  20            // padded LDS row stride (floats)

// ---------------------------------------------------------------------------
// Kernel 1: repack lora_B [E, O, R] -> BcatT [O, E*R] (row-major, like W)
// ---------------------------------------------------------------------------
__global__ void molex_prep(const float* __restrict__ loraB, float* __restrict__ BcatT) {
    int idx = blockIdx.x * 256 + threadIdx.x;      // 64*1024 total
    if (idx < ODIM * KEXT) {
        int o = idx >> 6;
        int j = idx & 63;                          // j = e*8 + r
        int e = j >> 3;
        int r = j & 7;
        BcatT[(size_t)o * KEXT + j] = loraB[(size_t)e * ODIM * RANK + (size_t)o * RANK + r];
    }
}

// ---------------------------------------------------------------------------
// Kernel 2: t1[N, 80] = x @ P^T, P = [lora_A (64 rows); gate_w (8); zeros (8)]
// WG = 128 tokens x 80 cols; 8 waves, each 16x80 (5 WMMA C-tiles).
// ---------------------------------------------------------------------------
__global__ __launch_bounds__(256) void molex_xa(const float* __restrict__ x,
                                                const float* __restrict__ loraA,
                                                const float* __restrict__ gate_w,
                                                float* __restrict__ t1) {
    __shared__ float lsA[128 * LDSP];
    __shared__ float lsB[T1W * LDSP];
    const int tid   = threadIdx.x;
    const int lane  = tid & 31;
    const int wv    = tid >> 5;
    const int mbase = blockIdx.x * 128;
    const int m_in  = lane & 15;
    const int khalf = lane >> 4;

    v8f acc[5] = {};

    for (int kb = 0; kb < DIM; kb += 16) {
#pragma unroll
        for (int t = 0; t < 2; ++t) {              // A tile: 128x16
            int id  = tid + t * 256;
            int row = id >> 2;
            int c4  = (id & 3) * 4;
            float4 av = *(const float4*)(x + (size_t)(mbase + row) * DIM + kb + c4);
            *(float4*)(&lsA[row * LDSP + c4]) = av;
        }
#pragma unroll
        for (int t = 0; t < 2; ++t) {              // B tile: 80x16
            int id = tid + t * 256;
            if (id < T1W * 4) {
                int row = id >> 2;
                int c4  = (id & 3) * 4;
                float4 bv;
                if (row < 64)      bv = *(const float4*)(loraA + (size_t)row * DIM + kb + c4);
                else if (row < 72) bv = *(const float4*)(gate_w + (size_t)(row - 64) * DIM + kb + c4);
                else               bv = make_float4(0.f, 0.f, 0.f, 0.f);
                *(float4*)(&lsB[row * LDSP + c4]) = bv;
            }
        }
        __syncthreads();
#pragma unroll
        for (int kk = 0; kk < 16; kk += 4) {
            const int k0 = kk + 2 * khalf;
            v2f af = *(const v2f*)(&lsA[(wv * 16 + m_in) * LDSP + k0]);
#pragma unroll
            for (int j = 0; j < 5; ++j) {
                v2f bf = *(const v2f*)(&lsB[(j * 16 + m_in) * LDSP + k0]);
                acc[j] = __builtin_amdgcn_wmma_f32_16x16x4_f32(
                    false, af, false, bf, (short)0, acc[j], false, false);
            }
        }
        __syncthreads();
    }
#pragma unroll
    for (int j = 0; j < 5; ++j) {
        int col   = j * 16 + m_in;
        int rbase = mbase + wv * 16 + 8 * khalf;
#pragma unroll
        for (int v = 0; v < 8; ++v)
            t1[(size_t)(rbase + v) * T1W + col] = acc[j][v];
    }
}

// ---------------------------------------------------------------------------
// Kernel 3: per-token top-2 + softmax; y[n, e*8+r] = gated, scaled xA.
// One wave per token; no dot products left here.
// ---------------------------------------------------------------------------
__global__ __launch_bounds__(256) void molex_topk(const float* __restrict__ t1,
                                                  const float* __restrict__ gate_b,
                                                  float* __restrict__ yv) {
    const int tid  = threadIdx.x;
    const int lane = tid & 31;
    const int wv   = tid >> 5;
    const int n    = blockIdx.x * 8 + wv;
    const float* tr = t1 + (size_t)n * T1W;

    float a0 = tr[lane];
    float a1 = tr[32 + lane];
    float l[8];
#pragma unroll
    for (int e = 0; e < NEXP; ++e) l[e] = tr[64 + e] + gate_b[e];   // broadcast loads

    int e1 = 0; float m1 = l[0];
#pragma unroll
    for (int e = 1; e < NEXP; ++e) if (l[e] > m1) { m1 = l[e]; e1 = e; }
    int e2 = -1; float m2 = -__builtin_huge_valf();
#pragma unroll
    for (int e = 0; e < NEXP; ++e) if (e != e1 && l[e] > m2) { m2 = l[e]; e2 = e; }
    float p2 = __expf(m2 - m1);
    float g1 = 1.f / (1.f + p2);
    float g2 = p2 * g1;

#pragma unroll
    for (int t = 0; t < 2; ++t) {
        int j = lane + t * 32;                     // 0..63
        int e = j >> 3;
        float coef = (e == e1) ? (SCALING * g1) : ((e == e2) ? (SCALING * g2) : 0.f);
        yv[(size_t)n * KEXT + j] = coef * (t ? a1 : a0);
    }
}

// ---------------------------------------------------------------------------
// Kernel 4: out = x @ W^T + y @ BcatT^T + bias  (K = 1024 + 64 = 1088)
// WG tile 128x128, 8 waves of 32x64 (2x4 WMMA C-tiles).
// Double-buffered LDS, register-staged fetch of chunk c+1 during compute of c.
// ---------------------------------------------------------------------------
__global__ __launch_bounds__(256) void molex_gemm(const float* __restrict__ x,
                                                  const float* __restrict__ W,
                                                  const float* __restrict__ bias,
                                                  const float* __restrict__ yv,
                                                  const float* __restrict__ BcatT,
                                                  float* __restrict__ out) {
    __shared__ float lsA[2][128 * LDSP];
    __shared__ float lsB[2][128 * LDSP];

    const int tid   = threadIdx.x;
    const int lane  = tid & 31;
    const int wv    = tid >> 5;
    const int wm    = wv & 3;                      // 4 wave-rows
    const int wn    = wv >> 2;                     // 2 wave-cols
    const int mbase = blockIdx.x * 128;
    const int nbase = blockIdx.y * 128;
    const int m_in  = lane & 15;
    const int khalf = lane >> 4;

    const int NCHUNK = (DIM + KEXT) / 16;          // 68

    v8f acc[2][4] = {};
    float4 areg[2], breg[2];

    auto fetch = [&](int c) {
        int kofs = c * 16;
        const float* Asrc; const float* Bsrc; int ld;
        if (kofs < DIM) { Asrc = x;  Bsrc = W;     ld = DIM; }
        else            { Asrc = yv; Bsrc = BcatT; ld = KEXT; kofs -= DIM; }
#pragma unroll
        for (int t = 0; t < 2; ++t) {
            int id  = tid + t * 256;               // 0..511
            int row = id >> 2;
            int c4  = (id & 3) * 4;
            areg[t] = *(const float4*)(Asrc + (size_t)(mbase + row) * ld + kofs + c4);
            breg[t] = *(const float4*)(Bsrc + (size_t)(nbase + row) * ld + kofs + c4);
        }
    };
    auto stage = [&](int buf) {
#pragma unroll
        for (int t = 0; t < 2; ++t) {
            int id  = tid + t * 256;
            int row = id >> 2;
            int c4  = (id & 3) * 4;
            *(float4*)(&lsA[buf][row * LDSP + c4]) = areg[t];
            *(float4*)(&lsB[buf][row * LDSP + c4]) = breg[t];
        }
    };

    fetch(0);
    stage(0);
    __syncthreads();

    for (int c = 0; c < NCHUNK; ++c) {
        const int buf = c & 1;
        if (c + 1 < NCHUNK) fetch(c + 1);          // overlap next-chunk loads

#pragma unroll
        for (int kk = 0; kk < 16; kk += 4) {
            const int k0 = kk + 2 * khalf;
            v2f af[2], bf[4];
#pragma unroll
            for (int i = 0; i < 2; ++i)
                af[i] = *(const v2f*)(&lsA[buf][(wm * 32 + i * 16 + m_in) * LDSP + k0]);
#pragma unroll
            for (int j = 0; j < 4; ++j)
                bf[j] = *(const v2f*)(&lsB[buf][(wn * 64 + j * 16 + m_in) * LDSP + k0]);
#pragma unroll
            for (int i = 0; i < 2; ++i)
#pragma unroll
                for (int j = 0; j < 4; ++j)
                    acc[i][j] = __builtin_amdgcn_wmma_f32_16x16x4_f32(
                        false, af[i], false, bf[j], (short)0, acc[i][j], false, false);
        }

        if (c + 1 < NCHUNK) stage(buf ^ 1);        // write other buffer
        __syncthreads();                           // one barrier per chunk
    }

    // epilogue: bias add + store (C layout: VGPR v -> M = v + 8*khalf, N = lane%16)
#pragma unroll
    for (int j = 0; j < 4; ++j) {
        int col = nbase + wn * 64 + j * 16 + m_in;
        float bvs = bias[col];
#pragma unroll
        for (int i = 0; i < 2; ++i) {
            int rbase = mbase + wm * 32 + i * 16 + 8 * khalf;
#pragma unroll
            for (int v = 0; v < 8; ++v) {
                out[(size_t)(rbase + v) * ODIM + col] = acc[i][j][v] + bvs;
            }
        }
    }
}

// ---------------------------------------------------------------------------
extern "C" void kernel_launch(void* const* d_in, const int* in_sizes, int n_in,
                              void* d_out, int out_size, void* d_ws, size_t ws_size,
                              hipStream_t stream) {
    const float* x      = (const float*)d_in[0];
    const float* W      = (const float*)d_in[1];
    const float* bias   = (const float*)d_in[2];
    const float* loraA  = (const float*)d_in[3];
    const float* loraB  = (const float*)d_in[4];
    const float* gate_w = (const float*)d_in[5];
    const float* gate_b = (const float*)d_in[6];
    float* out = (float*)d_out;

    float* BcatT = (float*)d_ws;                           // 64*1024 floats
    float* yv    = BcatT + (size_t)ODIM * KEXT;            // 16384*64 floats
    float* t1    = yv + (size_t)N_TOK * KEXT;              // 16384*80 floats

    molex_prep<<<(ODIM * KEXT + 255) / 256, 256, 0, stream>>>(loraB, BcatT);
    molex_xa<<<N_TOK / 128, 256, 0, stream>>>(x, loraA, gate_w, t1);
    molex_topk<<<N_TOK / 8, 256, 0, stream>>>(t1, gate_b, yv);
    molex_gemm<<<dim3(N_TOK / 128, ODIM / 128), 256, 0, stream>>>(x, W, bias, yv, BcatT, out);
}